// PitchPulseGAT_13709535608910
// MI455X (gfx1250) — compile-verified
//
#include <hip/hip_runtime.h>
#include <math.h>

#define NNODES 50000
#define NEDGES 800000
#define NETOT  850000   // + self loops
#define NGRAPH 256
#define H1     4
#define F1     128      // H1*32
#define NEG_SLOPE 0.2f

typedef __attribute__((ext_vector_type(2))) float v2f;
typedef __attribute__((ext_vector_type(8))) float v8f;

__device__ __forceinline__ float leakyf(float x){ return x > 0.f ? x : NEG_SLOPE * x; }
__device__ __forceinline__ float eluf(float x)  { return x > 0.f ? x : __expf(x) - 1.f; }

// order-preserving float atomic max (init value must be -inf)
__device__ __forceinline__ void atomicMaxF(float* addr, float v){
  if (v >= 0.f) atomicMax((int*)addr, __float_as_int(v));
  else          atomicMin((unsigned int*)addr, __float_as_uint(v));
}

__global__ void fill_f32(float* __restrict__ p, int n, float v){
  int i = blockIdx.x * blockDim.x + threadIdx.x;
  if (i < n) p[i] = v;
}

// ---- Layer 1: h1 = x @ W1 (K=2), alpha_src/dst per head via wave32 reduction ----
__global__ void node_l1(const float* __restrict__ x, const float* __restrict__ W1,
                        const float* __restrict__ aS, const float* __restrict__ aD,
                        float* __restrict__ h1, float* __restrict__ alS, float* __restrict__ alD){
  int n = blockIdx.x;           // one block (128 threads = 4 waves) per node
  int j = threadIdx.x;          // output channel 0..127
  float h = x[n*2 + 0] * W1[j] + x[n*2 + 1] * W1[F1 + j];
  h1[(size_t)n * F1 + j] = h;
  int head = j >> 5, c = j & 31;          // wave32: one wave == one head
  float s = h * aS[head*32 + c];
  float d = h * aD[head*32 + c];
  #pragma unroll
  for (int off = 16; off > 0; off >>= 1){
    s += __shfl_xor(s, off, 32);
    d += __shfl_xor(d, off, 32);
  }
  if (c == 0){ alS[n*H1 + head] = s; alD[n*H1 + head] = d; }
}

// ---- Edge pass 1: segment max of leaky-relu logits ----
template<int HEADS>
__global__ void edge_max(const int* __restrict__ ei, const float* __restrict__ alS,
                         const float* __restrict__ alD, float* __restrict__ m){
  int t = blockIdx.x * blockDim.x + threadIdx.x;
  if (t >= NETOT * HEADS) return;
  int e = t / HEADS, h = t % HEADS;
  int s = (e < NEDGES) ? ei[e]          : (e - NEDGES);
  int d = (e < NEDGES) ? ei[NEDGES + e] : (e - NEDGES);
  float l = leakyf(alS[s*HEADS + h] + alD[d*HEADS + h]);
  atomicMaxF(&m[d*HEADS + h], l);
}

// ---- Edge pass 2: e' = exp(logit - max), store per-edge, segment sum ----
template<int HEADS>
__global__ void edge_exp(const int* __restrict__ ei, const float* __restrict__ alS,
                         const float* __restrict__ alD, const float* __restrict__ m,
                         float* __restrict__ den, float* __restrict__ eexp){
  int t = blockIdx.x * blockDim.x + threadIdx.x;
  if (t >= NETOT * HEADS) return;
  int e = t / HEADS, h = t % HEADS;
  int s = (e < NEDGES) ? ei[e]          : (e - NEDGES);
  int d = (e < NEDGES) ? ei[NEDGES + e] : (e - NEDGES);
  float l = leakyf(alS[s*HEADS + h] + alD[d*HEADS + h]);
  float v = __expf(l - m[d*HEADS + h]);
  eexp[t] = v;
  atomicAdd(&den[d*HEADS + h], v);
}

// ---- Edge pass 3: one wave32 per (edge, head); lane = channel; scatter-add ----
template<int HEADS>
__global__ void edge_scatter(const int* __restrict__ ei, const float* __restrict__ eexp,
                             const float* __restrict__ den, const float* __restrict__ feat,
                             float* __restrict__ out){
  int t = blockIdx.x * blockDim.x + threadIdx.x;
  int w = t >> 5, lane = t & 31;
  if (w >= NETOT * HEADS) return;
  int e = w / HEADS, h = w % HEADS;
  int s = (e < NEDGES) ? ei[e]          : (e - NEDGES);
  int d = (e < NEDGES) ? ei[NEDGES + e] : (e - NEDGES);
  const int fdim = HEADS * 32;
  float att = eexp[w] / (den[d*HEADS + h] + 1e-16f);      // wave-uniform
  float v = att * feat[(size_t)s * fdim + h*32 + lane];
  atomicAdd(&out[(size_t)d * fdim + h*32 + lane], v);
}

// ---- x2 = elu(out1 + b1) ----
__global__ void bias_elu(const float* __restrict__ in, const float* __restrict__ b,
                         float* __restrict__ out, int total, int fmask){
  int i = blockIdx.x * blockDim.x + threadIdx.x;
  if (i >= total) return;
  out[i] = eluf(in[i] + b[i & fmask]);
}

// ---- Layer-2 GEMM via exact-f32 WMMA: h2[N,32] = x2[N,128] @ W2[128,32] ----
// W2 (16 KB) staged once per block in LDS; one wave per 16-row tile; two 16x16
// f32 accumulators cover the 32 output columns.
__global__ void gemm_wmma_f32(const float* __restrict__ A, const float* __restrict__ B,
                              float* __restrict__ D){
  __shared__ float Bs[F1 * 32];            // 16 KB of the WGP's 320 KB LDS
  for (int i = threadIdx.x; i < F1 * 32; i += 256) Bs[i] = B[i];
  __syncthreads();                          // before any wave-uniform exit

  int lane = threadIdx.x & 31;
  int wave = (blockIdx.x * blockDim.x + threadIdx.x) >> 5;
  int rowBase = wave * 16;
  if (rowBase >= NNODES) return;            // wave-uniform exit; EXEC stays all-1s
  int half = lane >> 4;                     // 0: K pair {k,k+1}; 1: {k+2,k+3}
  int lr   = lane & 15;
  const float* arow = A + (size_t)(rowBase + lr) * F1 + half * 2;
  v8f acc0 = {}; v8f acc1 = {};
  #pragma unroll 4
  for (int k = 0; k < F1; k += 4){
    int kk = k + half * 2;
    v2f a;  a.x  = arow[k];                a.y  = arow[k + 1];
    v2f b0; b0.x = Bs[kk*32 + lr];         b0.y = Bs[(kk+1)*32 + lr];
    v2f b1; b1.x = Bs[kk*32 + 16 + lr];    b1.y = Bs[(kk+1)*32 + 16 + lr];
    acc0 = __builtin_amdgcn_wmma_f32_16x16x4_f32(false, a, false, b0, (short)0, acc0, false, false);
    acc1 = __builtin_amdgcn_wmma_f32_16x16x4_f32(false, a, false, b1, (short)0, acc1, false, false);
  }
  #pragma unroll
  for (int r = 0; r < 8; ++r){
    int ro = rowBase + half * 8 + r;        // C/D layout: VGPR r -> M=r (lanes 0-15), M=r+8 (16-31)
    D[(size_t)ro * 32 + lr]      = acc0[r];
    D[(size_t)ro * 32 + 16 + lr] = acc1[r];
  }
}

// ---- alpha for layer 2 (H=1): one wave per node ----
__global__ void node_alpha2(const float* __restrict__ h2, const float* __restrict__ aS,
                            const float* __restrict__ aD, float* __restrict__ alS,
                            float* __restrict__ alD){
  int t = blockIdx.x * blockDim.x + threadIdx.x;
  int n = t >> 5, c = t & 31;
  if (n >= NNODES) return;
  float v = h2[(size_t)n * 32 + c];
  float s = v * aS[c], d = v * aD[c];
  #pragma unroll
  for (int off = 16; off > 0; off >>= 1){
    s += __shfl_xor(s, off, 32);
    d += __shfl_xor(d, off, 32);
  }
  if (c == 0){ alS[n] = s; alD[n] = d; }
}

// ---- pooling ----
__global__ void count_nodes(const int* __restrict__ batch, float* __restrict__ cnt){
  int n = blockIdx.x * blockDim.x + threadIdx.x;
  if (n >= NNODES) return;
  atomicAdd(&cnt[batch[n]], 1.0f);
}

__global__ void pool_kernel(const float* __restrict__ out2, const float* __restrict__ b2,
                            const int* __restrict__ batch, float* __restrict__ pool){
  int t = blockIdx.x * blockDim.x + threadIdx.x;
  if (t >= NNODES * 32) return;
  int n = t >> 5, c = t & 31;
  float v = eluf(out2[t] + b2[c]);
  atomicAdd(&pool[batch[n]*32 + c], v);
}

__global__ void final_kernel(const float* __restrict__ pool, const float* __restrict__ cnt,
                             const float* __restrict__ Wlin, const float* __restrict__ blin,
                             float* __restrict__ out){
  int g = blockIdx.x, c = threadIdx.x;     // 32 threads = one wave per graph
  float v = pool[g*32 + c] / fmaxf(cnt[g], 1.0f) * Wlin[c];
  #pragma unroll
  for (int off = 16; off > 0; off >>= 1) v += __shfl_xor(v, off, 32);
  if (c == 0) out[g] = v + blin[0];
}

static inline size_t algn(size_t x){ return (x + 255) & ~(size_t)255; }

extern "C" void kernel_launch(void* const* d_in, const int* in_sizes, int n_in,
                              void* d_out, int out_size, void* d_ws, size_t ws_size,
                              hipStream_t stream) {
  const float* x    = (const float*)d_in[0];
  const int*   ei   = (const int*)  d_in[1];
  const int*   batch= (const int*)  d_in[2];
  const float* W1   = (const float*)d_in[3];
  const float* aS1  = (const float*)d_in[4];
  const float* aD1  = (const float*)d_in[5];
  const float* b1   = (const float*)d_in[6];
  const float* W2   = (const float*)d_in[7];
  const float* aS2  = (const float*)d_in[8];
  const float* aD2  = (const float*)d_in[9];
  const float* b2   = (const float*)d_in[10];
  const float* Wlin = (const float*)d_in[11];
  const float* blin = (const float*)d_in[12];
  float* out = (float*)d_out;

  // carve workspace
  char* p = (char*)d_ws;
  auto carve = [&](size_t nfloats)->float*{ float* r = (float*)p; p += algn(nfloats * 4); return r; };
  float* h1    = carve((size_t)NNODES * F1);
  float* alS1  = carve((size_t)NNODES * H1);
  float* alD1  = carve((size_t)NNODES * H1);
  float* m1    = carve((size_t)NNODES * H1);
  float* den1  = carve((size_t)NNODES * H1);
  float* eexp1 = carve((size_t)NETOT * H1);
  float* out1  = carve((size_t)NNODES * F1);
  float* x2    = carve((size_t)NNODES * F1);
  float* h2    = carve((size_t)NNODES * 32);
  float* alS2  = carve((size_t)NNODES);
  float* alD2  = carve((size_t)NNODES);
  float* m2    = carve((size_t)NNODES);
  float* den2  = carve((size_t)NNODES);
  float* eexp2 = carve((size_t)NETOT);
  float* out2  = carve((size_t)NNODES * 32);
  float* pool  = carve((size_t)NGRAPH * 32);
  float* cnt   = carve((size_t)NGRAPH);

  const int B = 256;
  auto gb = [&](int n){ return (n + B - 1) / B; };

  // ---- init ----
  fill_f32<<<gb(NNODES*H1), B, 0, stream>>>(m1,   NNODES*H1, -INFINITY);
  fill_f32<<<gb(NNODES*H1), B, 0, stream>>>(den1, NNODES*H1, 0.f);
  fill_f32<<<gb(NNODES*F1), B, 0, stream>>>(out1, NNODES*F1, 0.f);
  fill_f32<<<gb(NNODES),    B, 0, stream>>>(m2,   NNODES,    -INFINITY);
  fill_f32<<<gb(NNODES),    B, 0, stream>>>(den2, NNODES,    0.f);
  fill_f32<<<gb(NNODES*32), B, 0, stream>>>(out2, NNODES*32, 0.f);
  fill_f32<<<gb(NGRAPH*32), B, 0, stream>>>(pool, NGRAPH*32, 0.f);
  fill_f32<<<gb(NGRAPH),    B, 0, stream>>>(cnt,  NGRAPH,    0.f);

  // ---- layer 1 ----
  node_l1<<<NNODES, 128, 0, stream>>>(x, W1, aS1, aD1, h1, alS1, alD1);
  edge_max<H1><<<gb(NETOT*H1), B, 0, stream>>>(ei, alS1, alD1, m1);
  edge_exp<H1><<<gb(NETOT*H1), B, 0, stream>>>(ei, alS1, alD1, m1, den1, eexp1);
  edge_scatter<H1><<<gb(NETOT*H1*32), B, 0, stream>>>(ei, eexp1, den1, h1, out1);
  bias_elu<<<gb(NNODES*F1), B, 0, stream>>>(out1, b1, x2, NNODES*F1, F1 - 1);

  // ---- layer 2: WMMA GEMM + attention (H=1) ----
  {
    int waves = NNODES / 16;                 // 3125, exact
    int blocks = (waves + 7) / 8;            // 8 waves per 256-thread block
    gemm_wmma_f32<<<blocks, 256, 0, stream>>>(x2, W2, h2);
  }
  node_alpha2<<<gb(NNODES*32), B, 0, stream>>>(h2, aS2, aD2, alS2, alD2);
  edge_max<1><<<gb(NETOT), B, 0, stream>>>(ei, alS2, alD2, m2);
  edge_exp<1><<<gb(NETOT), B, 0, stream>>>(ei, alS2, alD2, m2, den2, eexp2);
  edge_scatter<1><<<gb(NETOT*32), B, 0, stream>>>(ei, eexp2, den2, h2, out2);

  // ---- pool + linear ----
  count_nodes<<<gb(NNODES), B, 0, stream>>>(batch, cnt);
  pool_kernel<<<gb(NNODES*32), B, 0, stream>>>(out2, b2, batch, pool);
  final_kernel<<<NGRAPH, 32, 0, stream>>>(pool, cnt, Wlin, blin, out);
}